// TreeEncoder_16458314678339
// MI455X (gfx1250) — compile-verified
//
#include <hip/hip_runtime.h>

// ---------------------------------------------------------------------------
// TreeEncoder on gfx1250 (CDNA5): gather -> WMMA f32 GEMM -> quad pool
//   kernel 0: pack W into fragment-ready, K-padded, permuted layout (+zero row)
//   kernel 1: async-gather A tile to LDS (GLOBAL_LOAD_ASYNC_TO_LDS_B128),
//             16x16x4 f32 WMMA, 2 M-tiles per wave sharing each B fragment
//   kernel 2: mean pool over <=4 children
// ---------------------------------------------------------------------------

typedef __attribute__((ext_vector_type(2))) float v2f;
typedef __attribute__((ext_vector_type(8))) float v8f;
typedef int v4i __attribute__((vector_size(4 * sizeof(int))));
typedef __attribute__((address_space(1))) v4i glob_v4i;   // global int32x4
typedef __attribute__((address_space(3))) v4i lds_v4i;    // LDS int32x4

#define CH      64           // channels
#define KSLOT   9            // 3x3 neighbor slots
#define KDIM    585          // 9*(64+1) reduction dim
#define KPAD    588          // padded to x4 for 16x16x4 WMMA
#define KTILES  147          // KPAD/4
#define TILE_M  32           // fine nodes per workgroup (2 x 16-row WMMA tiles)
#define LDSROW  644          // LDS row stride (floats): %4==0 (16B rows), %64==4 (banks)
#define BLOCK   128          // 4 wave32 waves -> 4 x 16 output columns

#define WT_ELEMS (KTILES * CH * 4)   // 147*64*2*2 floats

// Permuted K order: j in [0,576): slot-major features (k*64+c) -> orig k*65+c
//                   j in [576,585): deltas                     -> orig k*65+64
__global__ __launch_bounds__(256)
void pack_w_kernel(const float* __restrict__ W,   // (585,64) row-major
                   float* __restrict__ Wt,        // (147,64,2,2) fragment layout
                   float* __restrict__ zeroRow)   // 64 zero floats
{
    const int i = blockIdx.x * 256 + threadIdx.x;
    if (i < WT_ELEMS) {
        const int r = i & 1;
        const int h = (i >> 1) & 1;
        const int n = (i >> 2) & 63;
        const int t = i >> 8;
        const int j = 4 * t + 2 * h + r;       // permuted K index
        float v = 0.0f;
        if (j < KDIM) {
            const int kk = (j < 576) ? ((j >> 6) * 65 + (j & 63))
                                     : ((j - 576) * 65 + 64);
            v = W[kk * CH + n];
        }
        Wt[i] = v;
    } else if (i < WT_ELEMS + CH) {
        zeroRow[i - WT_ELEMS] = 0.0f;
    }
}

__global__ __launch_bounds__(BLOCK)
void conv_wmma_kernel(const float* __restrict__ feats_fine,
                      const float* __restrict__ feats_coarse,
                      const float* __restrict__ Wt,      // fragment-ready W
                      const float* __restrict__ bias,    // (64,)
                      const float* __restrict__ zeroRow, // 256B of zeros
                      const int*   __restrict__ neigh_idx,
                      const int*   __restrict__ neigh_depth,
                      float*       __restrict__ conv,    // (N,64)
                      int N, int NP)
{
    extern __shared__ float G[];                         // [TILE_M][LDSROW]
    __shared__ const float* srcp[TILE_M * KSLOT];

    const int tid = threadIdx.x;
    const int nb  = blockIdx.x * TILE_M;

    // ---- Phase A: resolve per-slot source rows; write delta + K padding ----
    for (int t = tid; t < TILE_M * KSLOT; t += BLOCK) {
        const int nl   = t / KSLOT;
        const int k    = t % KSLOT;
        const int node = nb + nl;
        const float* src = zeroRow;                      // missing slot -> zeros
        float delta = 0.0f;
        if (node < N) {
            const int idx = neigh_idx[node * KSLOT + k];
            const int dep = neigh_depth[node * KSLOT + k];
            if (idx >= 0) {
                if (dep == 10) {
                    const int ci = idx < N - 1 ? idx : N - 1;
                    src = feats_fine + (long long)ci * CH;
                } else if (dep == 9) {
                    const int ci = idx < NP - 1 ? idx : NP - 1;
                    src = feats_coarse + (long long)ci * CH;
                    delta = 0.1f;                        // (10-9)/MAX_DEPTH
                }
            }
        }
        srcp[t] = src;
        G[nl * LDSROW + 576 + k] = delta;                // delta block
    }
    for (int t = tid; t < TILE_M * 3; t += BLOCK)        // K pad 585..587
        G[(t / 3) * LDSROW + KDIM + (t % 3)] = 0.0f;
    __syncthreads();

    // ---- Phase B: async Global->LDS gather (b128 granules, ASYNCcnt) ----
    for (int g = tid; g < TILE_M * KSLOT * (CH / 4); g += BLOCK) {
        const int nl = g / (KSLOT * (CH / 4));
        const int rr = g % (KSLOT * (CH / 4));
        const int k  = rr >> 4;
        const int c4 = rr & 15;
        const float* src4 = srcp[nl * KSLOT + k] + c4 * 4;
        float* dst = &G[nl * LDSROW + k * CH + c4 * 4];  // 16B aligned
#if defined(__has_builtin) && __has_builtin(__builtin_amdgcn_global_load_async_to_lds_b128)
        __builtin_amdgcn_global_load_async_to_lds_b128(
            (glob_v4i*)src4, (lds_v4i*)dst, 0, 0);
#else
        asm volatile("global_load_async_to_lds_b128 %0, %1, off"
                     :: "v"((unsigned)(unsigned long long)dst), "v"(src4)
                     : "memory");
#endif
    }
#if defined(__has_builtin) && __has_builtin(__builtin_amdgcn_s_wait_asynccnt)
    __builtin_amdgcn_s_wait_asynccnt(0);
#else
    asm volatile("s_wait_asynccnt 0" ::: "memory");
#endif
    __syncthreads();

    // ---- Phase C: branch-free K loop, 2 WMMAs per shared B fragment ----
    const int wave = tid >> 5;
    const int lane = tid & 31;
    const int m    = lane & 15;
    const int half = lane >> 4;
    const int col  = wave * 16 + m;

    v8f acc0, acc1;
    const float bn = bias[col];
    #pragma unroll
    for (int i = 0; i < 8; ++i) { acc0[i] = bn; acc1[i] = bn; }

    const v2f* wv  = (const v2f*)Wt + (col * 2 + half); // stride 128 v2f per tile
    const float* g0 = &G[m * LDSROW];
    const float* g1 = &G[(16 + m) * LDSROW];

    for (int t = 0; t < KTILES; ++t) {
        const int kb = 4 * t + 2 * half;
        const v2f a0 = *(const v2f*)(g0 + kb);
        const v2f a1 = *(const v2f*)(g1 + kb);
        const v2f bb = wv[(long long)t * 128];
        acc0 = __builtin_amdgcn_wmma_f32_16x16x4_f32(false, a0, false, bb,
                                                     (short)0, acc0, false, false);
        acc1 = __builtin_amdgcn_wmma_f32_16x16x4_f32(false, a1, false, bb,
                                                     (short)0, acc1, false, false);
    }

    // ---- Store D tiles ----
    #pragma unroll
    for (int v = 0; v < 8; ++v) {
        const int r0 = nb + v + half * 8;
        const int r1 = nb + 16 + v + half * 8;
        if (r0 < N) conv[(long long)r0 * CH + col] = acc0[v];
        if (r1 < N) conv[(long long)r1 * CH + col] = acc1[v];
    }
}

// ---------------------------------------------------------------------------
__global__ __launch_bounds__(256)
void pool_kernel(const float* __restrict__ conv,
                 const int*   __restrict__ children_idx,
                 float*       __restrict__ out,
                 int N, int NP)
{
    const int t = blockIdx.x * blockDim.x + threadIdx.x;
    if (t >= NP * (CH / 4)) return;
    const int p = t / (CH / 4);
    const int q = t % (CH / 4);

    float sx = 0.f, sy = 0.f, sz = 0.f, sw = 0.f;
    float cnt = 0.f;
    #pragma unroll
    for (int j = 0; j < 4; ++j) {
        const int ci = children_idx[p * 4 + j];
        if (ci >= 0) {
            const int cc = ci < N - 1 ? ci : N - 1;
            const float4 v =
                *reinterpret_cast<const float4*>(conv + (long long)cc * CH + q * 4);
            sx += v.x; sy += v.y; sz += v.z; sw += v.w;
            cnt += 1.f;
        }
    }
    const float inv = 1.0f / fmaxf(cnt, 1.0f);
    float4 r = make_float4(sx * inv, sy * inv, sz * inv, sw * inv);
    *reinterpret_cast<float4*>(out + (long long)p * CH + q * 4) = r;
}

// ---------------------------------------------------------------------------
extern "C" void kernel_launch(void* const* d_in, const int* in_sizes, int n_in,
                              void* d_out, int out_size, void* d_ws, size_t ws_size,
                              hipStream_t stream)
{
    const float* feats_fine   = (const float*)d_in[0];
    const float* feats_coarse = (const float*)d_in[1];
    const float* W            = (const float*)d_in[2];
    const float* bias         = (const float*)d_in[3];
    const int*   neigh_idx    = (const int*)d_in[4];
    const int*   neigh_depth  = (const int*)d_in[5];
    const int*   children_idx = (const int*)d_in[6];

    const int N  = in_sizes[0] / CH;
    const int NP = in_sizes[1] / CH;

    // d_ws layout: conv (N*64 f32) | Wt (147*64*4 f32) | zeroRow (64 f32)
    float* conv    = (float*)d_ws;
    float* Wt      = conv + (long long)N * CH;
    float* zeroRow = Wt + WT_ELEMS;
    float* out     = (float*)d_out;

    const int packTotal = WT_ELEMS + CH;
    pack_w_kernel<<<(packTotal + 255) / 256, 256, 0, stream>>>(W, Wt, zeroRow);

    const int grid1 = (N + TILE_M - 1) / TILE_M;
    const size_t ldsBytes = (size_t)TILE_M * LDSROW * sizeof(float);
    conv_wmma_kernel<<<grid1, BLOCK, ldsBytes, stream>>>(
        feats_fine, feats_coarse, Wt, bias, zeroRow,
        neigh_idx, neigh_depth, conv, N, NP);

    const int total2 = NP * (CH / 4);
    pool_kernel<<<(total2 + 255) / 256, 256, 0, stream>>>(
        conv, children_idx, out, N, NP);
}